// GMGN_35957466202791
// MI455X (gfx1250) — compile-verified
//
#include <hip/hip_runtime.h>

// Toeplitz matvec w[b,o] = sum_t Ah[(N-1)+t-o] * u[b,t] via fp32 WMMA 16x16x4.
// N=8193, batch=32 -> GEMM M=8193, N=32, K=8193 (4.3 GFLOP, WMMA-pipe bound).
// K split in two across blockIdx.y (partials + deterministic reduce) for
// occupancy; each wave computes two 16x16 row tiles sharing one B fragment
// (Toeplitz: tile-1 A window = tile-0 A window - 16). Loop kept in the
// "load-at-t + compiler unroll" shape that round 1 showed compiles to clauses
// of b64 loads with immediate offsets and split loadcnt waits.

#define NN 8193
#define AH_LEN (2 * NN - 1)       // 16385
#define PAD 128                   // zero padding before Ah (max row index 8319)
#define AH_TOTAL 16584            // PAD + AH_LEN + tail pad (max idx used 16515)
#define USTRIDE 8200              // padded u row stride (max offset read 8195)
#define BATCH 32
#define KSPLIT 4096               // K-half boundary

typedef __attribute__((ext_vector_type(2))) float v2f;
typedef __attribute__((ext_vector_type(8))) float v8f;

// ---- closed-form multigrid kernel construction -----------------------------
__device__ __forceinline__ float band_at(const float* __restrict__ b, int d) {
  // band endpoints (d = +/-7) are zeroed by the reference
  return (d >= -6 && d <= 6) ? b[d + 7] : 0.0f;
}

__device__ __forceinline__ float ah1(const float* __restrict__ ks,
                                     const float* __restrict__ b0, int k) {
  float v = (k & 1) ? 0.5f * (ks[(k - 1) >> 1] + ks[(k + 1) >> 1]) : ks[k >> 1];
  return v + band_at(b0, k - 2048);
}

__device__ __forceinline__ float ah2(const float* __restrict__ ks,
                                     const float* __restrict__ b0,
                                     const float* __restrict__ b1, int k) {
  float v = (k & 1) ? 0.5f * (ah1(ks, b0, (k - 1) >> 1) + ah1(ks, b0, (k + 1) >> 1))
                    : ah1(ks, b0, k >> 1);
  return v + band_at(b1, k - 4096);
}

__device__ __forceinline__ float ah3(const float* __restrict__ ks,
                                     const float* __restrict__ b0,
                                     const float* __restrict__ b1,
                                     const float* __restrict__ b2, int k) {
  float v = (k & 1) ? 0.5f * (ah2(ks, b0, b1, (k - 1) >> 1) + ah2(ks, b0, b1, (k + 1) >> 1))
                    : ah2(ks, b0, b1, k >> 1);
  return v + band_at(b2, k - 8192);
}

__global__ void build_ahpad(const float* __restrict__ ks, const float* __restrict__ b0,
                            const float* __restrict__ b1, const float* __restrict__ b2,
                            float* __restrict__ out) {
  int i = blockIdx.x * blockDim.x + threadIdx.x;
  if (i >= AH_TOTAL) return;
  int k = i - PAD;
  out[i] = (k >= 0 && k < AH_LEN) ? ah3(ks, b0, b1, b2, k) : 0.0f;
}

__global__ void pack_u(const float* __restrict__ u, float* __restrict__ upad) {
  int i = blockIdx.x * blockDim.x + threadIdx.x;
  if (i >= BATCH * USTRIDE) return;
  int b = i / USTRIDE;
  int t = i - b * USTRIDE;
  upad[i] = (t < NN) ? u[b * NN + t] : 0.0f;   // zero tail: K-loop overrun safe
}

// ---- main GEMM: fp32 WMMA 16x16x4, K-split, 2 row tiles per wave -----------
// Block = 256 threads = 8 waves: wave w -> col tile (w&1), row group (w>>1).
// Each wave: rows [o0,o0+16) and [o0+16,o0+32) x 16 batch cols; B frag shared.
__global__ void __launch_bounds__(256)
toep_wmma(const float* __restrict__ Ahpad, const float* __restrict__ upad,
          float* __restrict__ part) {
  const int lane = threadIdx.x & 31;
  const int wv   = threadIdx.x >> 5;
  const int ct   = wv & 1;           // batch-col tile 0..1
  const int rg   = wv >> 1;          // row group 0..3
  const int o0   = blockIdx.x * 128 + rg * 32;
  const int nlo  = lane & 15;        // = m for A-frags, = n for B/D-frags
  const int h    = lane >> 4;        // half-wave selects K pair {2h, 2h+1}

  const int t0   = blockIdx.y * KSPLIT;            // [0,4096) / [4096,8193)
  const int tend = (blockIdx.y == 0) ? KSPLIT : NN;

  // A[m,k] = Ah[(N-1)+(t+k)-(o+m)]: shifted windows; tile 1 is tile 0 - 16.
  const float* aptr0 = Ahpad + (PAD + (NN - 1) + 2 * h - (o0 + nlo));
  const float* aptr1 = aptr0 - 16;
  // B[k,n] = u[b0+n][t+k]  (8B-aligned: USTRIDE, t, 2h all even)
  const float* bptr  = upad + ((ct * 16 + nlo) * USTRIDE + 2 * h);

  v8f acc0 = 0.0f, acc1 = 0.0f;
#pragma unroll 8
  for (int t = t0; t < tend; t += 4) {
    v2f a0, a1, bv;
    a0.x = aptr0[t]; a0.y = aptr0[t + 1];         // merges into global_load_b64
    a1.x = aptr1[t]; a1.y = aptr1[t + 1];         // (offset -64B immediate)
    bv = *(const v2f*)(bptr + t);                 // aligned global_load_b64
    acc0 = __builtin_amdgcn_wmma_f32_16x16x4_f32(false, a0, false, bv,
                                                 (short)0, acc0, false, false);
    acc1 = __builtin_amdgcn_wmma_f32_16x16x4_f32(false, a1, false, bv,
                                                 (short)0, acc1, false, false);
  }

  // D layout: lane -> n = nlo, VGPR v -> m = v + 8h.  Partials per K-half.
  float* pout = part + (size_t)blockIdx.y * (BATCH * NN) + (size_t)(ct * 16 + nlo) * NN;
  const int ob = o0 + 8 * h;
  if (o0 + 32 <= NN) {               // wave-uniform: all but the last row-block
#pragma unroll
    for (int v = 0; v < 8; ++v) {
      pout[ob + v]      = acc0[v];
      pout[ob + 16 + v] = acc1[v];
    }
  } else {
#pragma unroll
    for (int v = 0; v < 8; ++v) {
      if (ob + v < NN)      pout[ob + v]      = acc0[v];
      if (ob + 16 + v < NN) pout[ob + 16 + v] = acc1[v];
    }
  }
}

__global__ void reduce_parts(const float* __restrict__ part, float* __restrict__ out) {
  int i = blockIdx.x * blockDim.x + threadIdx.x;
  if (i < BATCH * NN) out[i] = part[i] + part[BATCH * NN + i];
}

extern "C" void kernel_launch(void* const* d_in, const int* in_sizes, int n_in,
                              void* d_out, int out_size, void* d_ws, size_t ws_size,
                              hipStream_t stream) {
  (void)in_sizes; (void)n_in; (void)out_size; (void)ws_size;
  const float* u  = (const float*)d_in[0];
  // d_in[1] = x : unused by the reference computation
  const float* ks = (const float*)d_in[2];
  const float* b0 = (const float*)d_in[3];
  const float* b1 = (const float*)d_in[4];
  const float* b2 = (const float*)d_in[5];

  float* ws    = (float*)d_ws;
  float* Ahpad = ws;                              // AH_TOTAL floats
  float* upad  = Ahpad + AH_TOTAL;                // BATCH*USTRIDE floats
  float* part  = upad + (BATCH * USTRIDE);        // 2*BATCH*NN floats (~2.1 MB)

  build_ahpad<<<(AH_TOTAL + 255) / 256, 256, 0, stream>>>(ks, b0, b1, b2, Ahpad);
  pack_u<<<(BATCH * USTRIDE + 255) / 256, 256, 0, stream>>>(u, upad);

  dim3 grid((NN + 127) / 128, 2);                 // 65 row-blocks x 2 K-halves
  toep_wmma<<<grid, 256, 0, stream>>>(Ahpad, upad, part);

  reduce_parts<<<(BATCH * NN + 255) / 256, 256, 0, stream>>>(part, (float*)d_out);
}